// TemporalAggregator_19859928777381
// MI455X (gfx1250) — compile-verified
//
#include <hip/hip_runtime.h>
#include <math.h>

#define ALPHA_C 0.5f
#define BETA_C 0.9f

typedef __attribute__((ext_vector_type(2))) float v2f;
typedef __attribute__((ext_vector_type(8))) float v8f;

// ---------------- reductions (wave32) ----------------
__device__ __forceinline__ float warpSum(float v) {
#pragma unroll
  for (int o = 16; o > 0; o >>= 1) v += __shfl_xor(v, o, 32);
  return v;
}
__device__ __forceinline__ float warpMax(float v) {
#pragma unroll
  for (int o = 16; o > 0; o >>= 1) v = fmaxf(v, __shfl_xor(v, o, 32));
  return v;
}
__device__ __forceinline__ float blockSum(float v, float* s) {
  int lane = threadIdx.x & 31, wid = threadIdx.x >> 5;
  int nw = (blockDim.x + 31) >> 5;
  v = warpSum(v);
  __syncthreads();
  if (lane == 0) s[wid] = v;
  __syncthreads();
  if (wid == 0) {
    float r = (lane < nw) ? s[lane] : 0.0f;
    r = warpSum(r);
    if (lane == 0) s[0] = r;
  }
  __syncthreads();
  return s[0];
}
__device__ __forceinline__ float blockMax(float v, float* s) {
  int lane = threadIdx.x & 31, wid = threadIdx.x >> 5;
  int nw = (blockDim.x + 31) >> 5;
  v = warpMax(v);
  __syncthreads();
  if (lane == 0) s[wid] = v;
  __syncthreads();
  if (wid == 0) {
    float r = (lane < nw) ? s[lane] : -3.0e38f;
    r = warpMax(r);
    if (lane == 0) s[0] = r;
  }
  __syncthreads();
  return s[0];
}

// ---------------- pass 1: masked mean pool + t_max ----------------
// grid = B, block = D (256). One streaming pass over x[b] (coalesced).
__global__ void stats_kernel(const float* __restrict__ x, const float* __restrict__ ts,
                             const unsigned char* __restrict__ pad,
                             float* __restrict__ q, float* __restrict__ tmax_out,
                             int T, int D) {
  int b = blockIdx.x;
  int tid = threadIdx.x;  // == d
  __shared__ float s_valid[512];
  __shared__ float s_red[32];
  float tmx = -1.0e18f, vc = 0.0f;
  for (int t = tid; t < T; t += blockDim.x) {
    float v = (pad[(size_t)b * T + t] == 0) ? 1.0f : 0.0f;
    s_valid[t] = v;
    float tv = ts[(size_t)b * T + t];
    if (v != 0.0f) { tmx = fmaxf(tmx, tv); vc += 1.0f; }
  }
  __syncthreads();
  float tmax_b = blockMax(tmx, s_red);
  float vcnt = fmaxf(blockSum(vc, s_red), 1.0f);
  const float* xb = x + (size_t)b * T * D;
  float acc = 0.0f;
  for (int t = 0; t < T; ++t) acc += s_valid[t] * xb[(size_t)t * D + tid];
  q[(size_t)b * D + tid] = acc / vcnt;
  if (tid == 0) tmax_out[b] = tmax_b;
}

// ---------------- WMMA f32 GEMM: C[M,N] = A[M,K] * B ----------------
// BT=true : B[k][n] = Bm[n*K + k]  (i.e. C = A @ Bm^T)
// BT=false: B[k][n] = Bm[k*N + n]  (i.e. C = A @ Bm)
// One 16x16 tile per wave; K-loop of V_WMMA_F32_16X16X4_F32.
// VGPR layouts per ISA 7.12.2: A lanes 0-15 hold M rows, half selects K pair;
// C/D: VGPR i -> row tm + i + 8*half, col tn + (lane&15).
template <bool BT>
__global__ void wmma_gemm_f32_k(const float* __restrict__ A, const float* __restrict__ Bm,
                                float* __restrict__ C, int M, int N, int K) {
  int waveId = (int)blockIdx.x * (blockDim.x >> 5) + (threadIdx.x >> 5);
  int lane = threadIdx.x & 31;
  int tilesN = N >> 4;
  int tiles = (M >> 4) * tilesN;
  if (waveId >= tiles) return;  // wave-uniform: EXEC stays all-ones for WMMA
  int tm = (waveId / tilesN) << 4;
  int tn = (waveId % tilesN) << 4;
  int half = lane >> 4;
  int l16 = lane & 15;
  const float* arow = A + (size_t)(tm + l16) * K;
  v8f acc = {};
  for (int k = 0; k < K; k += 4) {
    v2f a, bv;
    int kk = k + (half << 1);
    a.x = arow[kk];
    a.y = arow[kk + 1];
    if (BT) {
      const float* brow = Bm + (size_t)(tn + l16) * K;
      bv.x = brow[kk];
      bv.y = brow[kk + 1];
    } else {
      bv.x = Bm[(size_t)kk * N + tn + l16];
      bv.y = Bm[(size_t)(kk + 1) * N + tn + l16];
    }
    acc = __builtin_amdgcn_wmma_f32_16x16x4_f32(false, a, false, bv, (short)0, acc,
                                                false, false);
  }
  float* crow = C + (size_t)(tm + (half << 3)) * N + tn + l16;
#pragma unroll
  for (int i = 0; i < 8; ++i) crow[(size_t)i * N] = acc[i];
}

// ---------------- pass 2: scores -> softmax*decay -> weighted pool ----------------
// grid = B, block = D (256 = 8 waves). x[b] (512 KB) is swept twice back-to-back
// by the same block -> second sweep hits L2.
__global__ void attn_kernel(const float* __restrict__ x, const float* __restrict__ ts,
                            const unsigned char* __restrict__ pad,
                            const float* __restrict__ cvec, const float* __restrict__ tmax_arr,
                            float* __restrict__ xw, int T, int D) {
  int b = blockIdx.x;
  int tid = threadIdx.x;
  int lane = tid & 31, wid = tid >> 5;
  __shared__ float s_sc[512];
  __shared__ float s_c[256];
  __shared__ float s_red[32];
  s_c[tid] = cvec[(size_t)b * D + tid];
  __syncthreads();
  const float* xb = x + (size_t)b * T * D;
  int nwaves = blockDim.x >> 5;
  // scores[t] = (x[b,t] . c[b]) / sqrt(D); one row per wave, lanes over D (coalesced)
  for (int t = wid; t < T; t += nwaves) {
    float acc = 0.0f;
    const float* row = xb + (size_t)t * D;
    for (int d = lane; d < D; d += 32) acc += row[d] * s_c[d];
    acc = warpSum(acc);
    if (lane == 0) s_sc[t] = acc * 0.0625f;  // 1/sqrt(256)
  }
  __syncthreads();
  float tmax_b = tmax_arr[b];
  float m = -3.0e38f;
  for (int t = tid; t < T; t += blockDim.x) {
    float sc = pad[(size_t)b * T + t] ? -1.0e9f : s_sc[t];
    s_sc[t] = sc;
    m = fmaxf(m, sc);
  }
  float smax = blockMax(m, s_red);  // contains barriers
  // u_t = lam_t * exp(sc-smax); w_t = u_t / (U + 1e-8 * Z)  (== ref's two-stage norm)
  float Z = 0.0f, U = 0.0f;
  for (int t = tid; t < T; t += blockDim.x) {
    float sc = s_sc[t];
    float e = expf(sc - smax);
    float valid = pad[(size_t)b * T + t] ? 0.0f : 1.0f;
    float dt = fmaxf(tmax_b - ts[(size_t)b * T + t], 0.0f) * (1.0f / 86400.0f);
    float lam = expf(-ALPHA_C * dt) * valid;
    float u = lam * e;
    s_sc[t] = u;
    Z += e;
    U += u;
  }
  float Zt = blockSum(Z, s_red);   // barriers also publish s_sc writes
  float Ut = blockSum(U, s_red);
  float inv = 1.0f / (Ut + 1.0e-8f * Zt);
  // xw[b,d] = sum_t w_t * x[b,t,d]; thread-per-d, coalesced rows
  float acc = 0.0f;
  for (int t = 0; t < T; ++t) acc += s_sc[t] * xb[(size_t)t * D + tid];
  xw[(size_t)b * D + tid] = acc * inv;
}

// ---------------- final: delta, M_t, T_event, p ----------------
__global__ void final_kernel(const float* __restrict__ L, const float* __restrict__ prevL,
                             const float* __restrict__ prevM,
                             const float* __restrict__ cls_w, const float* __restrict__ cls_b,
                             float* __restrict__ p_out, float* __restrict__ Tev,
                             float* __restrict__ M_out, int D) {
  int b = blockIdx.x;
  int d = threadIdx.x;  // blockDim == D
  __shared__ float s_red[32];
  __shared__ float s_m;
  float l = L[(size_t)b * D + d];
  float dl = l - prevL[(size_t)b * D + d];
  float ss = blockSum(dl * dl, s_red);
  if (d == 0) {
    float Mt = BETA_C * prevM[b] + (1.0f - BETA_C) * sqrtf(ss);
    s_m = Mt;
    M_out[b] = Mt;
  }
  __syncthreads();
  float Mt = s_m;
  size_t tb = (size_t)b * (2 * D + 1);
  Tev[tb + d] = l;
  Tev[tb + D + d] = dl;
  if (d == 0) Tev[tb + 2 * D] = Mt;
  float contrib = l * cls_w[d] + dl * cls_w[D + d];
  float ps = blockSum(contrib, s_red);
  if (d == 0) p_out[b] = ps + Mt * cls_w[2 * D] + cls_b[0];
}

extern "C" void kernel_launch(void* const* d_in, const int* in_sizes, int n_in,
                              void* d_out, int out_size, void* d_ws, size_t ws_size,
                              hipStream_t stream) {
  const float* x = (const float*)d_in[0];          // [B,T,D]
  const float* ts = (const float*)d_in[1];         // [B,T]
  const float* prevL = (const float*)d_in[2];      // [B,D]
  const float* prevM = (const float*)d_in[3];      // [B,1]
  const unsigned char* pad = (const unsigned char*)d_in[4];  // [B,T] bool
  const float* W_Q = (const float*)d_in[5];        // [D,D]
  const float* W_K = (const float*)d_in[6];        // [D,D]
  const float* W_V = (const float*)d_in[7];        // [D,D]
  const float* cls_w = (const float*)d_in[8];      // [1,2D+1]
  const float* cls_b = (const float*)d_in[9];      // [1]

  int B = in_sizes[3];            // prev_M is [B,1]
  int D = in_sizes[2] / B;        // prev_L is [B,D]
  int T = in_sizes[1] / B;        // timestamps is [B,T]

  // workspace layout (floats): q | Q | c | xw | tmax
  float* ws = (float*)d_ws;
  float* q = ws;
  float* Qb = q + (size_t)B * D;
  float* cb = Qb + (size_t)B * D;
  float* xw = cb + (size_t)B * D;
  float* tmax = xw + (size_t)B * D;

  // output layout: p[B] | T_event[B,2D+1] | L_t[B,D] | M_t[B]
  float* out = (float*)d_out;
  float* p_out = out;
  float* Tev = out + B;
  float* Lt = Tev + (size_t)B * (2 * D + 1);
  float* Mt = Lt + (size_t)B * D;

  stats_kernel<<<B, D, 0, stream>>>(x, ts, pad, q, tmax, T, D);

  int tiles = (B / 16) * (D / 16);
  int gemmBlocks = (tiles + 7) / 8;  // 8 waves (16x16 tiles) per 256-thread block
  // Q = q @ W_Q^T
  wmma_gemm_f32_k<true><<<gemmBlocks, 256, 0, stream>>>(q, W_Q, Qb, B, D, D);
  // c = Q @ W_K   (folds K-projection into the score vector)
  wmma_gemm_f32_k<false><<<gemmBlocks, 256, 0, stream>>>(Qb, W_K, cb, B, D, D);

  attn_kernel<<<B, D, 0, stream>>>(x, ts, pad, cb, tmax, xw, T, D);

  // L_t = xw @ W_V^T, written directly into output slot
  wmma_gemm_f32_k<true><<<gemmBlocks, 256, 0, stream>>>(xw, W_V, Lt, B, D, D);

  final_kernel<<<B, D, 0, stream>>>(Lt, prevL, prevM, cls_w, cls_b, p_out, Tev, Mt, D);
}